// ODC_50663434224361
// MI455X (gfx1250) — compile-verified
//
#include <hip/hip_runtime.h>
#include <hip/hip_bf16.h>

typedef __attribute__((ext_vector_type(2))) float v2f;
typedef __attribute__((ext_vector_type(4))) float v4f;
typedef __attribute__((ext_vector_type(8))) float v8f;

#define ODC_MOMENTUM 0.5f
#define ODC_EPS 1e-10f

// ---------------------------------------------------------------------------
// Kernel 1: non-temporal float4 copy of the bank (1 GB of streaming traffic;
// NT keeps it from thrashing the 192MB L2), labels int->float, counter init.
// ---------------------------------------------------------------------------
__global__ void odc_copy_kernel(const float* __restrict__ bank,
                                const int* __restrict__ labels,
                                float* __restrict__ out_bank,
                                float* __restrict__ out_labels,
                                long long nbank, int L,
                                int* __restrict__ counter) {
    if (blockIdx.x == 0 && threadIdx.x == 0) *counter = 0;
    long long i = (long long)blockIdx.x * blockDim.x + threadIdx.x;
    long long stride = (long long)gridDim.x * blockDim.x;
    const v4f* src = (const v4f*)bank;
    v4f* dst = (v4f*)out_bank;
    long long n4 = nbank >> 2;
    for (long long k = i; k < n4; k += stride) {
        v4f v = __builtin_nontemporal_load(&src[k]);
        __builtin_nontemporal_store(v, &dst[k]);
    }
    for (long long k = i; k < (long long)L; k += stride)
        out_labels[k] = (float)labels[k];
}

// ---------------------------------------------------------------------------
// Kernel 2a (D == 256 fast path): one wave per batch row, 8 floats per lane,
// __shfl_xor butterfly norms -- no block barriers, fully coalesced b128s.
// ---------------------------------------------------------------------------
__global__ void odc_update_wave_kernel(const float* __restrict__ bank,
                                       const float* __restrict__ feature,
                                       const int* __restrict__ ind,
                                       float* __restrict__ out_bank,
                                       float* __restrict__ feat_new,
                                       int B) {
    const int tid = threadIdx.x;
    const int wave = tid >> 5;
    const int lane = tid & 31;
    const int b = blockIdx.x * 8 + wave;
    if (b >= B) return;
    const int D = 256;

    const v4f* frow = (const v4f*)(feature + (long long)b * D);
    v4f x0 = frow[lane * 2];
    v4f x1 = frow[lane * 2 + 1];

    float s = x0.x*x0.x + x0.y*x0.y + x0.z*x0.z + x0.w*x0.w
            + x1.x*x1.x + x1.y*x1.y + x1.z*x1.z + x1.w*x1.w;
    for (int off = 16; off > 0; off >>= 1) s += __shfl_xor(s, off, 32);
    float inv1 = 1.0f / (sqrtf(s) + ODC_EPS);

    const long long row = (long long)ind[b];
    const v4f* orow = (const v4f*)(bank + row * D);
    v4f o0 = orow[lane * 2];
    v4f o1 = orow[lane * 2 + 1];

    v4f y0, y1;
    y0.x = (1.0f-ODC_MOMENTUM)*o0.x + ODC_MOMENTUM*x0.x*inv1;
    y0.y = (1.0f-ODC_MOMENTUM)*o0.y + ODC_MOMENTUM*x0.y*inv1;
    y0.z = (1.0f-ODC_MOMENTUM)*o0.z + ODC_MOMENTUM*x0.z*inv1;
    y0.w = (1.0f-ODC_MOMENTUM)*o0.w + ODC_MOMENTUM*x0.w*inv1;
    y1.x = (1.0f-ODC_MOMENTUM)*o1.x + ODC_MOMENTUM*x1.x*inv1;
    y1.y = (1.0f-ODC_MOMENTUM)*o1.y + ODC_MOMENTUM*x1.y*inv1;
    y1.z = (1.0f-ODC_MOMENTUM)*o1.z + ODC_MOMENTUM*x1.z*inv1;
    y1.w = (1.0f-ODC_MOMENTUM)*o1.w + ODC_MOMENTUM*x1.w*inv1;

    float s2 = y0.x*y0.x + y0.y*y0.y + y0.z*y0.z + y0.w*y0.w
             + y1.x*y1.x + y1.y*y1.y + y1.z*y1.z + y1.w*y1.w;
    for (int off = 16; off > 0; off >>= 1) s2 += __shfl_xor(s2, off, 32);
    float inv2 = 1.0f / (sqrtf(s2) + ODC_EPS);

    y0.x *= inv2; y0.y *= inv2; y0.z *= inv2; y0.w *= inv2;
    y1.x *= inv2; y1.y *= inv2; y1.z *= inv2; y1.w *= inv2;

    v4f* wrow = (v4f*)(feat_new + (long long)b * D);
    wrow[lane * 2]     = y0;
    wrow[lane * 2 + 1] = y1;
    v4f* brow = (v4f*)(out_bank + row * D);
    brow[lane * 2]     = y0;
    brow[lane * 2 + 1] = y1;
}

// ---------------------------------------------------------------------------
// Kernel 2b (generic D fallback): block-per-row with LDS tree reductions.
// ---------------------------------------------------------------------------
__global__ void odc_update_kernel(const float* __restrict__ bank,
                                  const float* __restrict__ feature,
                                  const int* __restrict__ ind,
                                  float* __restrict__ out_bank,
                                  float* __restrict__ feat_new,
                                  int D) {
    __shared__ float sred[1024];
    const int b = blockIdx.x;
    const int t = threadIdx.x;
    const long long row = (long long)ind[b];

    float f = feature[(long long)b * D + t];
    sred[t] = f * f;
    __syncthreads();
    for (int s = blockDim.x >> 1; s > 0; s >>= 1) {
        if (t < s) sred[t] += sred[t + s];
        __syncthreads();
    }
    float n1 = sred[0];
    __syncthreads();

    float fn = f / (sqrtf(n1) + ODC_EPS);
    float oldv = bank[row * D + t];
    float x = (1.0f - ODC_MOMENTUM) * oldv + ODC_MOMENTUM * fn;

    sred[t] = x * x;
    __syncthreads();
    for (int s = blockDim.x >> 1; s > 0; s >>= 1) {
        if (t < s) sred[t] += sred[t + s];
        __syncthreads();
    }
    float n2 = sred[0];

    float y = x / (sqrtf(n2) + ODC_EPS);
    feat_new[(long long)b * D + t] = y;
    out_bank[row * D + t] = y;
}

// ---------------------------------------------------------------------------
// Kernel 3: centroids @ feat_new^T fused with argmax over C via
// V_WMMA_F32_16X16X4_F32 (exact fp32).
//  * feat tile staged via GLOBAL_LOAD_ASYNC_TO_LDS_B128 (ASYNCcnt, no VGPR
//    round trip), padded row stride 260 (260%64==4 -> conflict-free b64s).
//  * Two centroid tiles per wave -> two independent accumulator chains
//    sharing one B fragment: 2x matrix-pipe ILP, half the LDS reads.
// ---------------------------------------------------------------------------
#define FT_STRIDE 260

__launch_bounds__(256)
__global__ void odc_sim_argmax_kernel(const float* __restrict__ centroids,
                                      const float* __restrict__ feat_new,
                                      const int* __restrict__ ind,
                                      const int* __restrict__ label_bank,
                                      float* __restrict__ out_labels,
                                      int* __restrict__ counter,
                                      int C, int D) {
    __shared__ float ftile[16 * FT_STRIDE];
    __shared__ float sV[8][16];
    __shared__ int   sI[8][16];

    const int tid  = threadIdx.x;
    const int wave = tid >> 5;
    const int lane = tid & 31;
    const int lane16 = lane & 15;
    const int half = lane >> 4;            // 0: K=0,1   1: K=2,3
    const int bt = blockIdx.x;             // batch tile (16 columns)

    // Async global->LDS staging of feat_new tile [16][D], padded rows.
    // Low 32 bits of the generic __shared__ pointer == LDS byte offset.
    {
        const unsigned lds_base = (unsigned)(uintptr_t)(void*)ftile;
        const int q = D >> 2;              // float4s per row
        for (int i = tid; i < 16 * q; i += 256) {   // uniform trip count
            int r = i / q;
            int c4 = i - r * q;
            const float* g = feat_new + ((long long)(bt * 16 + r)) * D + c4 * 4;
            unsigned loff = lds_base + (unsigned)(r * FT_STRIDE + c4 * 4) * 4u;
            asm volatile("global_load_async_to_lds_b128 %0, %1, off"
                         :: "v"(loff), "v"(g) : "memory");
        }
        asm volatile("s_wait_asynccnt 0x0" ::: "memory");
    }
    __syncthreads();

    float bestV = -3.402823466e+38f;
    int   bestI = 0;

    const float* fbase = ftile + lane16 * FT_STRIDE + 2 * half;
    const int C16 = C >> 4;

    // Wave w covers tiles {w, w+8, ..., w+56}; process them in pairs
    // (j, j+8) with two independent accumulators sharing the B fragment.
    for (int j = wave; j < C16; j += 16) {       // uniform per wave
        const int mtA = j, mtB = j + 8;
        v8f acc0 = {}, acc1 = {};
        const float* aA = centroids + ((long long)(mtA * 16 + lane16)) * D + 2 * half;
        const float* aB = centroids + ((long long)(mtB * 16 + lane16)) * D + 2 * half;
        for (int k0 = 0; k0 < D; k0 += 4) {
            v2f bfrag = *(const v2f*)(fbase + k0);   // shared B: 4x16 from LDS
            v2f a0 = *(const v2f*)(aA + k0);
            v2f a1 = *(const v2f*)(aB + k0);
            acc0 = __builtin_amdgcn_wmma_f32_16x16x4_f32(
                false, a0, false, bfrag, (short)0, acc0, false, false);
            acc1 = __builtin_amdgcn_wmma_f32_16x16x4_f32(
                false, a1, false, bfrag, (short)0, acc1, false, false);
        }
        // D layout: lane<16 -> rows tile*16 + r ; lane>=16 -> tile*16 + 8 + r
        const int mbase0 = mtA * 16 + 8 * half;
        const int mbase1 = mtB * 16 + 8 * half;
#pragma unroll
        for (int r = 0; r < 8; ++r) {
            float v = acc0[r];
            int m = mbase0 + r;
            if (v > bestV || (v == bestV && m < bestI)) { bestV = v; bestI = m; }
        }
#pragma unroll
        for (int r = 0; r < 8; ++r) {
            float v = acc1[r];
            int m = mbase1 + r;
            if (v > bestV || (v == bestV && m < bestI)) { bestV = v; bestI = m; }
        }
    }

    // Combine lane l with lane l+16 (same output column n = lane16).
    {
        float oV = __shfl_down(bestV, 16, 32);
        int   oI = __shfl_down(bestI, 16, 32);
        if (oV > bestV || (oV == bestV && oI < bestI)) { bestV = oV; bestI = oI; }
    }
    if (lane < 16) { sV[wave][lane] = bestV; sI[wave][lane] = bestI; }
    __syncthreads();

    // Cross-wave argmax + label scatter + change count.
    if (tid < 16) {
        float bv = sV[0][tid];
        int   bi = sI[0][tid];
        for (int w = 1; w < 8; ++w) {
            float v = sV[w][tid];
            int   m = sI[w][tid];
            if (v > bv || (v == bv && m < bi)) { bv = v; bi = m; }
        }
        int b = bt * 16 + tid;
        int row = ind[b];
        out_labels[row] = (float)bi;
        if (bi != label_bank[row]) atomicAdd(counter, 1);
    }
}

// ---------------------------------------------------------------------------
// Kernel 4: change_ratio = counter / B
// ---------------------------------------------------------------------------
__global__ void odc_finalize_kernel(const int* __restrict__ counter,
                                    float* __restrict__ out_ratio, int B) {
    *out_ratio = (float)(*counter) / (float)B;
}

// ---------------------------------------------------------------------------
extern "C" void kernel_launch(void* const* d_in, const int* in_sizes, int n_in,
                              void* d_out, int out_size, void* d_ws, size_t ws_size,
                              hipStream_t stream) {
    const float* bank      = (const float*)d_in[0];   // [L, D]
    const int*   labels    = (const int*)d_in[1];     // [L]
    const float* centroids = (const float*)d_in[2];   // [C, D]
    const int*   ind       = (const int*)d_in[3];     // [B]
    const float* feature   = (const float*)d_in[4];   // [B, D]

    const long long bankN = (long long)in_sizes[0];   // L*D
    const int L = in_sizes[1];
    const int B = in_sizes[3];
    const int D = (int)(bankN / L);                   // 256
    const int C = in_sizes[2] / D;                    // 1024

    float* out_bank   = (float*)d_out;                // [L*D]
    float* out_labels = out_bank + bankN;             // [L]
    float* out_ratio  = out_labels + L;               // [1]

    float* feat_new = (float*)d_ws;                   // [B*D] scratch
    int* counter = (int*)((char*)d_ws + (size_t)B * (size_t)D * sizeof(float));

    odc_copy_kernel<<<4096, 256, 0, stream>>>(bank, labels, out_bank,
                                              out_labels, bankN, L, counter);
    if (D == 256) {
        odc_update_wave_kernel<<<(B + 7) / 8, 256, 0, stream>>>(
            bank, feature, ind, out_bank, feat_new, B);
    } else {
        odc_update_kernel<<<B, D, 0, stream>>>(bank, feature, ind, out_bank,
                                               feat_new, D);
    }
    odc_sim_argmax_kernel<<<B / 16, 256, 0, stream>>>(centroids, feat_new, ind,
                                                      labels, out_labels,
                                                      counter, C, D);
    odc_finalize_kernel<<<1, 1, 0, stream>>>(counter, out_ratio, B);
}